// Net_all_1735166787790
// MI455X (gfx1250) — compile-verified
//
#include <hip/hip_runtime.h>
#include <cstdint>
#include <cstddef>

#define DIM 128
#define HCC 72
#define NHEAD 9
#define CHAN 8
#define NEG_SLOPE 0.2f

typedef __attribute__((ext_vector_type(2))) float v2f;
typedef __attribute__((ext_vector_type(8))) float v8f;

// ---------------------------------------------------------------- fill
__global__ void fill_kernel(float* __restrict__ p, float v, long long n) {
  long long i = (long long)blockIdx.x * blockDim.x + threadIdx.x;
  if (i < n) p[i] = v;
}

// ---------------------------------------------------------------- GEMM h = x @ W  (M=N nodes, K=128, N=72) via WMMA f32 16x16x4
// One wave per 16x16 output tile. Out-of-range lanes clamp their load
// addresses to valid memory: a garbage A-row m only corrupts D-row m and a
// garbage B-col n only corrupts D-col n, and those are never stored, so no
// per-iteration predication is needed -> branch-free inner loop.
__global__ __launch_bounds__(32) void gemm_wmma_kernel(const float* __restrict__ x,
                                                       const float* __restrict__ W,
                                                       float* __restrict__ h,
                                                       int nrows) {
  const int rt = blockIdx.x << 4;      // row tile base (M)
  const int ct = blockIdx.y << 4;      // col tile base (N of GEMM, <=72)
  const int lane = threadIdx.x & 31;
  const int half = lane >> 4;          // 0: lanes 0-15, 1: lanes 16-31
  const int l = lane & 15;

  const int arow = min(rt + l, nrows - 1);       // clamped A row for this lane
  const int bcol_real = ct + l;                  // real B/D column
  const int bcol = min(bcol_real, HCC - 1);      // clamped B column

  // A 16x4 layout: lane half 0 holds K=k0,k0+1; half 1 holds K=k0+2,k0+3.
  // B 4x16 layout: rows split across half-waves like C/D.
  const float* ap = x + (size_t)arow * DIM + 2 * half;
  const float* bp = W + (size_t)(2 * half) * HCC + bcol;

  v8f acc = {};
#pragma unroll 8
  for (int k0 = 0; k0 < DIM; k0 += 4) {
    const float2 av = *reinterpret_cast<const float2*>(ap);
    v2f a; a.x = av.x; a.y = av.y;
    v2f b;
    b.x = bp[0];
    b.y = bp[HCC];
    acc = __builtin_amdgcn_wmma_f32_16x16x4_f32(
        /*neg_a=*/false, a, /*neg_b=*/false, b,
        /*c_mod=*/(short)0, acc, /*reuse_a=*/false, /*reuse_b=*/false);
    ap += 4;
    bp += 4 * HCC;
  }

  // D layout: VGPR j -> row rt+j (lanes 0-15) / rt+8+j (lanes 16-31), col = ct+l
  if (bcol_real < HCC) {
#pragma unroll
    for (int j = 0; j < 8; ++j) {
      const int row = rt + j + 8 * half;
      if (row < nrows) h[(size_t)row * HCC + bcol_real] = acc[j];
    }
  }
}

// ---------------------------------------------------------------- per-node attention projections a_s, a_d  [N,9]
__global__ void attn_proj_kernel(const float* __restrict__ h,
                                 const float* __restrict__ att_src,
                                 const float* __restrict__ att_dst,
                                 float* __restrict__ a_s,
                                 float* __restrict__ a_d, int n) {
  const int node = blockIdx.x * blockDim.x + threadIdx.x;
  if (node >= n) return;
  const float* hp = h + (size_t)node * HCC;
#pragma unroll
  for (int hd = 0; hd < NHEAD; ++hd) {
    float s = 0.0f, d = 0.0f;
#pragma unroll
    for (int c = 0; c < CHAN; ++c) {
      const float v = hp[hd * CHAN + c];
      s += v * att_src[hd * CHAN + c];
      d += v * att_dst[hd * CHAN + c];
    }
    a_s[(size_t)node * NHEAD + hd] = s;
    a_d[(size_t)node * NHEAD + hd] = d;
  }
}

__device__ __forceinline__ void edge_endpoints(const long long* __restrict__ ei,
                                               long long E, long long e,
                                               int& s, int& d) {
  if (e < E) { s = (int)ei[e]; d = (int)ei[E + e]; }
  else       { s = (int)(e - E); d = s; }          // appended self-loops
}

__device__ __forceinline__ float lrelu(float v) {
  return v > 0.0f ? v : NEG_SLOPE * v;
}

// ---------------------------------------------------------------- pass 1: per-dst, per-head running max (stable softmax)
__global__ void edge_max_kernel(const long long* __restrict__ ei,
                                const float* __restrict__ a_s,
                                const float* __restrict__ a_d,
                                float* __restrict__ m,
                                long long E, int n) {
  const long long idx = (long long)blockIdx.x * blockDim.x + threadIdx.x;
  const long long total = (E + n) * NHEAD;
  if (idx >= total) return;
  const long long e = idx / NHEAD;
  const int hd = (int)(idx % NHEAD);
  int s, d; edge_endpoints(ei, E, e, s, d);
  const float v = lrelu(a_s[(size_t)s * NHEAD + hd] + a_d[(size_t)d * NHEAD + hd]);
  atomicMax(&m[(size_t)d * NHEAD + hd], v);
}

// ---------------------------------------------------------------- pass 2: denom = sum exp(score - max)
__global__ void edge_expsum_kernel(const long long* __restrict__ ei,
                                   const float* __restrict__ a_s,
                                   const float* __restrict__ a_d,
                                   const float* __restrict__ m,
                                   float* __restrict__ denom,
                                   long long E, int n) {
  const long long idx = (long long)blockIdx.x * blockDim.x + threadIdx.x;
  const long long total = (E + n) * NHEAD;
  if (idx >= total) return;
  const long long e = idx / NHEAD;
  const int hd = (int)(idx % NHEAD);
  int s, d; edge_endpoints(ei, E, e, s, d);
  const float v = lrelu(a_s[(size_t)s * NHEAD + hd] + a_d[(size_t)d * NHEAD + hd]);
  const float ex = expf(v - m[(size_t)d * NHEAD + hd]);
  atomicAdd(&denom[(size_t)d * NHEAD + hd], ex);
}

// ---------------------------------------------------------------- pass 3: weighted scatter-add of h[src] into o[dst]
__global__ void edge_aggregate_kernel(const long long* __restrict__ ei,
                                      const float* __restrict__ a_s,
                                      const float* __restrict__ a_d,
                                      const float* __restrict__ m,
                                      const float* __restrict__ denom,
                                      const float* __restrict__ h,
                                      float* __restrict__ o,
                                      long long E, int n) {
  const long long idx = (long long)blockIdx.x * blockDim.x + threadIdx.x;
  const long long total = (E + n) * NHEAD;
  if (idx >= total) return;
  const long long e = idx / NHEAD;
  const int hd = (int)(idx % NHEAD);
  int s, d; edge_endpoints(ei, E, e, s, d);
  const float v = lrelu(a_s[(size_t)s * NHEAD + hd] + a_d[(size_t)d * NHEAD + hd]);
  const float w = expf(v - m[(size_t)d * NHEAD + hd]) /
                  (denom[(size_t)d * NHEAD + hd] + 1e-16f);
  const float4* hp = reinterpret_cast<const float4*>(h + (size_t)s * HCC + hd * CHAN); // 32B aligned
  const float4 h0 = hp[0];
  const float4 h1 = hp[1];
  float* op = o + (size_t)d * HCC + hd * CHAN;
  atomicAdd(op + 0, h0.x * w);
  atomicAdd(op + 1, h0.y * w);
  atomicAdd(op + 2, h0.z * w);
  atomicAdd(op + 3, h0.w * w);
  atomicAdd(op + 4, h1.x * w);
  atomicAdd(op + 5, h1.y * w);
  atomicAdd(op + 6, h1.z * w);
  atomicAdd(op + 7, h1.w * w);
}

// ---------------------------------------------------------------- bias + relu + concat + FNN + softmax
__global__ void finalize_kernel(const float* __restrict__ o,       // [3][N][72]
                                const float* __restrict__ b1,
                                const float* __restrict__ b2,
                                const float* __restrict__ b3,
                                const float* __restrict__ fW,      // [216,2]
                                const float* __restrict__ fb,      // [2]
                                float* __restrict__ out, int n) {
  const int node = blockIdx.x * blockDim.x + threadIdx.x;
  if (node >= n) return;
  float l0 = fb[0], l1 = fb[1];
  const float* biases[3] = {b1, b2, b3};
  for (int g = 0; g < 3; ++g) {
    const float* og = o + ((size_t)g * n + node) * HCC;
    const float* bg = biases[g];
#pragma unroll 8
    for (int c = 0; c < HCC; ++c) {
      float v = og[c] + bg[c];
      v = v > 0.0f ? v : 0.0f;
      const int r = g * HCC + c;
      l0 += v * fW[r * 2 + 0];
      l1 += v * fW[r * 2 + 1];
    }
  }
  const float mx = fmaxf(l0, l1);
  const float e0 = expf(l0 - mx);
  const float e1 = expf(l1 - mx);
  const float inv = 1.0f / (e0 + e1);
  out[(size_t)node * 2 + 0] = e0 * inv;
  out[(size_t)node * 2 + 1] = e1 * inv;
}

// ================================================================ host
static inline size_t align256(size_t x) { return (x + 255) & ~(size_t)255; }

extern "C" void kernel_launch(void* const* d_in, const int* in_sizes, int n_in,
                              void* d_out, int out_size, void* d_ws, size_t ws_size,
                              hipStream_t stream) {
  // Input dict order: per graph g: x, edge_index, W, att_src, att_dst, b ; then fnn_W, fnn_b
  const float*     x[3]  = {(const float*)d_in[0],  (const float*)d_in[6],  (const float*)d_in[12]};
  const long long* ei[3] = {(const long long*)d_in[1], (const long long*)d_in[7], (const long long*)d_in[13]};
  const float*     Wm[3] = {(const float*)d_in[2],  (const float*)d_in[8],  (const float*)d_in[14]};
  const float*     as_[3]= {(const float*)d_in[3],  (const float*)d_in[9],  (const float*)d_in[15]};
  const float*     ad_[3]= {(const float*)d_in[4],  (const float*)d_in[10], (const float*)d_in[16]};
  const float*     bv[3] = {(const float*)d_in[5],  (const float*)d_in[11], (const float*)d_in[17]};
  const float*     fW    = (const float*)d_in[18];
  const float*     fb    = (const float*)d_in[19];
  float*           out   = (float*)d_out;

  const int       N = in_sizes[0] / DIM;        // 50000
  const long long E = in_sizes[1] / 2;          // 1.6M

  // Workspace carve-out
  char* ws = (char*)d_ws;
  size_t off = 0;
  float* o     = (float*)(ws + off); off = align256(off + (size_t)3 * N * HCC * sizeof(float));
  float* h     = (float*)(ws + off); off = align256(off + (size_t)N * HCC * sizeof(float));
  float* a_s   = (float*)(ws + off); off = align256(off + (size_t)N * NHEAD * sizeof(float));
  float* a_d   = (float*)(ws + off); off = align256(off + (size_t)N * NHEAD * sizeof(float));
  float* mbuf  = (float*)(ws + off); off = align256(off + (size_t)N * NHEAD * sizeof(float));
  float* dnm   = (float*)(ws + off); off = align256(off + (size_t)N * NHEAD * sizeof(float));
  (void)ws_size; (void)n_in; (void)out_size;

  const int TPB = 256;
  const long long oTot = (long long)3 * N * HCC;
  fill_kernel<<<(unsigned)((oTot + TPB - 1) / TPB), TPB, 0, stream>>>(o, 0.0f, oTot);

  const long long nodeHead = (long long)N * NHEAD;
  const long long edgeHead = (E + N) * (long long)NHEAD;
  const unsigned ehBlocks  = (unsigned)((edgeHead + TPB - 1) / TPB);
  const unsigned nhBlocks  = (unsigned)((nodeHead + TPB - 1) / TPB);
  const unsigned nBlocks   = (unsigned)((N + TPB - 1) / TPB);

  dim3 gemmGrid((N + 15) / 16, (HCC + 15) / 16);   // 3125 x 5 waves

  for (int g = 0; g < 3; ++g) {
    gemm_wmma_kernel<<<gemmGrid, 32, 0, stream>>>(x[g], Wm[g], h, N);
    attn_proj_kernel<<<nBlocks, TPB, 0, stream>>>(h, as_[g], ad_[g], a_s, a_d, N);
    fill_kernel<<<nhBlocks, TPB, 0, stream>>>(mbuf, -__builtin_huge_valf(), nodeHead);
    fill_kernel<<<nhBlocks, TPB, 0, stream>>>(dnm, 0.0f, nodeHead);
    edge_max_kernel<<<ehBlocks, TPB, 0, stream>>>(ei[g], a_s, a_d, mbuf, E, N);
    edge_expsum_kernel<<<ehBlocks, TPB, 0, stream>>>(ei[g], a_s, a_d, mbuf, dnm, E, N);
    edge_aggregate_kernel<<<ehBlocks, TPB, 0, stream>>>(ei[g], a_s, a_d, mbuf, dnm, h,
                                                        o + (size_t)g * N * HCC, E, N);
  }
  finalize_kernel<<<nBlocks, TPB, 0, stream>>>(o, bv[0], bv[1], bv[2], fW, fb, out, N);
}